// Distiller_67577015435275
// MI455X (gfx1250) — compile-verified
//
#include <hip/hip_runtime.h>
#include <math.h>

typedef __attribute__((ext_vector_type(2))) float v2f;
typedef __attribute__((ext_vector_type(8))) float v8f;

#define EPS 1e-12f
#define KC 32
#define SA2 144   // A LDS row stride in float2 units (288 floats == 32 mod 64 banks)
#define SB2 80    // B LDS row stride in float2 units (160 floats == 32 mod 64 banks)

// ---------------------------------------------------------------------------
// Kernel 1: Fc[b,c] = sum_{spatial} |F[b,c,:]|  for TF and SF. One wave per row.
// ---------------------------------------------------------------------------
__global__ __launch_bounds__(256) void fc_kernel(const float* __restrict__ TF,
                                                 const float* __restrict__ SF,
                                                 float* __restrict__ FcT,
                                                 float* __restrict__ FcS) {
    int row  = blockIdx.x * 8 + (threadIdx.x >> 5);   // 0..16383 (8192 T rows, 8192 S rows)
    int lane = threadIdx.x & 31;
    const float* src = (row < 8192) ? TF : SF;
    float*       dst = (row < 8192) ? FcT : FcS;
    int r = row & 8191;
    const float* p = src + (size_t)r * 1024;
    float s = 0.0f;
    #pragma unroll
    for (int i = 0; i < 32; ++i) s += fabsf(p[lane + i * 32]);
    #pragma unroll
    for (int m = 16; m; m >>= 1) s += __shfl_xor(s, m, 32);
    if (lane == 0) dst[r] = s;
}

// ---------------------------------------------------------------------------
// Kernel 2: fused dual Gram GEMM via V_WMMA_F32_16X16X4_F32.
//   ATT_T[m,n] = sum_c (Fc_T[c]*TF[c,m]) * TF[c,n]   (same for S)
// Workgroup: 128m x 64n; 8 waves; each wave = 16m x 64n for both T and S
// (8 v8f accumulators). LDS holds K-pairs as float2 so each WMMA operand is
// one aligned ds_load_b64.
// Per-row ssS/ssT/sst partials accumulated via atomics across n-tiles.
// ---------------------------------------------------------------------------
__global__ __launch_bounds__(256) void sa_gemm_kernel(const float* __restrict__ TF,
                                                      const float* __restrict__ SF,
                                                      const float* __restrict__ FcT,
                                                      const float* __restrict__ FcS,
                                                      float* __restrict__ ssS,
                                                      float* __restrict__ ssT,
                                                      float* __restrict__ sst) {
    __shared__ v2f lds2[2 * 16 * SA2 + 2 * 16 * SB2];   // 7168 float2 = 56 KB
    v2f* aT2 = lds2;
    v2f* aS2 = lds2 + 16 * SA2;
    v2f* bT2 = lds2 + 2 * 16 * SA2;
    v2f* bS2 = lds2 + 2 * 16 * SA2 + 16 * SB2;

    const int b  = blockIdx.z;
    const int m0 = blockIdx.y * 128;
    const int n0 = blockIdx.x * 64;
    const int tid  = threadIdx.x;
    const int wv   = tid >> 5;       // wave id 0..7 -> owns rows m0 + wv*16 ..
    const int lane = tid & 31;
    const int lsub = lane & 15;
    const int hi   = lane >> 4;      // half-wave: K-pair select (K=0,1 vs K=2,3)

    const float* Tb  = TF + (size_t)b * 512 * 1024;
    const float* Sb  = SF + (size_t)b * 512 * 1024;
    const float* fcT = FcT + b * 512;
    const float* fcS = FcS + b * 512;

    v8f accT[4] = {{}, {}, {}, {}};
    v8f accS[4] = {{}, {}, {}, {}};

    const int am = wv * 16 + lsub;

    for (int kc = 0; kc < 512; kc += KC) {
        __syncthreads();
        // Stage A: KC/2=16 k-pair rows x 128 m, scaled by Fc (T and S)
        for (int f = tid; f < 16 * 128; f += 256) {
            int k2 = f >> 7;
            int m  = f & 127;
            int k  = kc + 2 * k2;
            size_t r0 = (size_t)k * 1024;
            v2f t, s;
            t.x = Tb[r0 + m0 + m] * fcT[k];
            t.y = Tb[r0 + 1024 + m0 + m] * fcT[k + 1];
            s.x = Sb[r0 + m0 + m] * fcS[k];
            s.y = Sb[r0 + 1024 + m0 + m] * fcS[k + 1];
            aT2[k2 * SA2 + m] = t;
            aS2[k2 * SA2 + m] = s;
        }
        // Stage B: 16 k-pair rows x 64 n, unscaled (T and S)
        for (int f = tid; f < 16 * 64; f += 256) {
            int k2 = f >> 6;
            int n  = f & 63;
            int k  = kc + 2 * k2;
            size_t r0 = (size_t)k * 1024;
            v2f t, s;
            t.x = Tb[r0 + n0 + n];
            t.y = Tb[r0 + 1024 + n0 + n];
            s.x = Sb[r0 + n0 + n];
            s.y = Sb[r0 + 1024 + n0 + n];
            bT2[k2 * SB2 + n] = t;
            bS2[k2 * SB2 + n] = s;
        }
        __syncthreads();
        #pragma unroll
        for (int k4 = 0; k4 < KC; k4 += 4) {
            const int krow = (k4 >> 1) + hi;            // float2 row: K = k4+2*hi, k4+2*hi+1
            v2f a_t = aT2[krow * SA2 + am];
            v2f a_s = aS2[krow * SA2 + am];
            #pragma unroll
            for (int j = 0; j < 4; ++j) {
                v2f b_t = bT2[krow * SB2 + j * 16 + lsub];
                v2f b_s = bS2[krow * SB2 + j * 16 + lsub];
                accT[j] = __builtin_amdgcn_wmma_f32_16x16x4_f32(false, a_t, false, b_t,
                                                                (short)0, accT[j], false, false);
                accS[j] = __builtin_amdgcn_wmma_f32_16x16x4_f32(false, a_s, false, b_s,
                                                                (short)0, accS[j], false, false);
            }
        }
    }

    // Per-row partial sums over this wave's 64 columns, then 16-lane reduce.
    #pragma unroll
    for (int i = 0; i < 8; ++i) {
        float pss = 0.f, ptt = 0.f, pst = 0.f;
        #pragma unroll
        for (int j = 0; j < 4; ++j) {
            float sv = accS[j][i], tv = accT[j][i];
            pss += sv * sv;
            ptt += tv * tv;
            pst += sv * tv;
        }
        #pragma unroll
        for (int m = 1; m < 16; m <<= 1) {
            pss += __shfl_xor(pss, m, 32);
            ptt += __shfl_xor(ptt, m, 32);
            pst += __shfl_xor(pst, m, 32);
        }
        if (lsub == 0) {
            int row = m0 + wv * 16 + i + 8 * hi;   // C-layout: VGPR i -> M = i + 8*hi
            atomicAdd(&ssS[b * 1024 + row], pss);
            atomicAdd(&ssT[b * 1024 + row], ptt);
            atomicAdd(&sst[b * 1024 + row], pst);
        }
    }
}

// ---------------------------------------------------------------------------
// Kernel 3: IC loss per batch. 21x21 grams over K=16384, LDS-chunked.
// ---------------------------------------------------------------------------
#define CC 21
#define MM 16384
__global__ __launch_bounds__(256) void ic_kernel(const float* __restrict__ t_out,
                                                 const float* __restrict__ s_out,
                                                 float* __restrict__ ic_part) {
    __shared__ float sBuf[CC * 128];
    __shared__ float tBuf[CC * 128];
    __shared__ float Gs[CC * CC];
    __shared__ float Gt[CC * CC];
    __shared__ float nS[CC], nT[CC];
    __shared__ float red[256];

    const int b = blockIdx.x;
    const int tid = threadIdx.x;
    const float* S = s_out + (size_t)b * CC * MM;
    const float* T = t_out + (size_t)b * CC * MM;

    const int p0 = tid, p1 = tid + 256;                 // pair ids (i*21+j), 441 total
    float aS0 = 0.f, aT0 = 0.f, aS1 = 0.f, aT1 = 0.f;

    for (int mc = 0; mc < MM; mc += 128) {
        __syncthreads();
        for (int f = tid; f < CC * 128; f += 256) {
            int i = f >> 7, m = f & 127;
            sBuf[f] = S[(size_t)i * MM + mc + m];
            tBuf[f] = T[(size_t)i * MM + mc + m];
        }
        __syncthreads();
        {
            int i = p0 / CC, j = p0 % CC;
            float as = 0.f, at = 0.f;
            for (int m = 0; m < 128; ++m) {
                as += sBuf[i * 128 + m] * sBuf[j * 128 + m];
                at += tBuf[i * 128 + m] * tBuf[j * 128 + m];
            }
            aS0 += as; aT0 += at;
        }
        if (p1 < CC * CC) {
            int i = p1 / CC, j = p1 % CC;
            float as = 0.f, at = 0.f;
            for (int m = 0; m < 128; ++m) {
                as += sBuf[i * 128 + m] * sBuf[j * 128 + m];
                at += tBuf[i * 128 + m] * tBuf[j * 128 + m];
            }
            aS1 += as; aT1 += at;
        }
    }
    __syncthreads();
    if (p0 < CC * CC) { Gs[p0] = aS0; Gt[p0] = aT0; }
    if (p1 < CC * CC) { Gs[p1] = aS1; Gt[p1] = aT1; }
    __syncthreads();
    if (tid < CC) {
        float ss = 0.f, tt = 0.f;
        for (int j = 0; j < CC; ++j) {
            ss += Gs[tid * CC + j] * Gs[tid * CC + j];
            tt += Gt[tid * CC + j] * Gt[tid * CC + j];
        }
        nS[tid] = fmaxf(sqrtf(ss), EPS);
        nT[tid] = fmaxf(sqrtf(tt), EPS);
    }
    __syncthreads();
    float part = 0.f;
    for (int p = tid; p < CC * CC; p += 256) {
        int i = p / CC;
        float d = Gs[p] / nS[i] - Gt[p] / nT[i];
        part += d * d;
    }
    red[tid] = part;
    __syncthreads();
    for (int k = 128; k; k >>= 1) { if (tid < k) red[tid] += red[tid + k]; __syncthreads(); }
    if (tid == 0) ic_part[b] = red[0];
}

// ---------------------------------------------------------------------------
// Kernel 4: finalize both scalars.
//   SA row contribution: ssS/nS^2 + ssT/nT^2 - 2*sst/(nS*nT), nX = max(sqrt(ssX), EPS)
// ---------------------------------------------------------------------------
__global__ __launch_bounds__(256) void finalize_kernel(const float* __restrict__ ssS,
                                                       const float* __restrict__ ssT,
                                                       const float* __restrict__ sst,
                                                       const float* __restrict__ ic_part,
                                                       float* __restrict__ out_ic,
                                                       float* __restrict__ out_sa) {
    __shared__ float red[256];
    const int tid = threadIdx.x;
    float s = 0.f;
    for (int r = tid; r < 16 * 1024; r += 256) {
        float a = ssS[r], t = ssT[r], x = sst[r];
        float nSv = fmaxf(sqrtf(a), EPS);
        float nTv = fmaxf(sqrtf(t), EPS);
        s += a / (nSv * nSv) + t / (nTv * nTv) - 2.0f * x / (nSv * nTv);
    }
    red[tid] = s;
    __syncthreads();
    for (int k = 128; k; k >>= 1) { if (tid < k) red[tid] += red[tid + k]; __syncthreads(); }
    if (tid == 0) {
        *out_sa = red[0] * (1.0f / (16.0f * 1024.0f * 1024.0f));   // SA_LAMBDA=1
        float ic = 0.f;
        for (int bb = 0; bb < 16; ++bb) ic += ic_part[bb];
        *out_ic = ic * (1.0f / (21.0f * 16.0f));                   // IC_LAMBDA=1
    }
}

// ---------------------------------------------------------------------------
// Workspace layout (floats):
//   [0      .. 8191 ]  FcT
//   [8192   .. 16383]  FcS
//   [16384  .. 32767]  ssS   (zeroed each launch)
//   [32768  .. 49151]  ssT   (zeroed)
//   [49152  .. 65535]  sst   (zeroed)
//   [65536  .. 65551]  ic_part
// ---------------------------------------------------------------------------
extern "C" void kernel_launch(void* const* d_in, const int* in_sizes, int n_in,
                              void* d_out, int out_size, void* d_ws, size_t ws_size,
                              hipStream_t stream) {
    const float* TF    = (const float*)d_in[0];
    const float* SF    = (const float*)d_in[1];
    const float* t_out = (const float*)d_in[2];
    const float* s_out = (const float*)d_in[3];
    float* out = (float*)d_out;
    float* ws  = (float*)d_ws;

    float* FcT     = ws;
    float* FcS     = ws + 8192;
    float* ssS     = ws + 16384;
    float* ssT     = ws + 32768;
    float* sstv    = ws + 49152;
    float* ic_part = ws + 65536;

    const size_t sout_elems = (size_t)16 * 21 * 128 * 128;   // 5,505,024

    // zero accumulators (capturable async ops only)
    hipMemsetAsync(ssS, 0, (3 * 16384 + 16) * sizeof(float), stream);
    // output 0: s_out passthrough
    hipMemcpyAsync(out, s_out, sout_elems * sizeof(float), hipMemcpyDeviceToDevice, stream);

    fc_kernel<<<2048, 256, 0, stream>>>(TF, SF, FcT, FcS);
    sa_gemm_kernel<<<dim3(16, 8, 16), 256, 0, stream>>>(TF, SF, FcT, FcS, ssS, ssT, sstv);
    ic_kernel<<<16, 256, 0, stream>>>(t_out, s_out, ic_part);
    finalize_kernel<<<1, 256, 0, stream>>>(ssS, ssT, sstv, ic_part,
                                           out + sout_elems, out + sout_elems + 1);
}